// IDEStateSpaceModel_27453430956254
// MI455X (gfx1250) — compile-verified
//
#include <hip/hip_runtime.h>

#define SNUM 64
#define DDIM 128
#define BATCH 16
#define TSTEPS 128
#define NTH 256

typedef __attribute__((ext_vector_type(2))) float v2f;
typedef __attribute__((ext_vector_type(8))) float v8f;

// -------------------------------------------------------------------------
// Kernel 0: shared geometry. lon/lat are broadcast across the batch in the
// reference, so h[i][j][2] is identical for every batch -> compute once.
// -------------------------------------------------------------------------
__global__ void ide_h_kernel(const float* __restrict__ lon,
                             const float* __restrict__ lat,
                             float* __restrict__ h_out) {
  __shared__ float cx[SNUM], cy[SNUM];
  __shared__ float lon0, lat0, iscale;
  const int tid = threadIdx.x;
  if (tid == 0) {
    float sl = 0.f, st = 0.f;
    for (int i = 0; i < SNUM; ++i) { sl += lon[i]; st += lat[i]; }
    lon0 = sl * (1.0f / SNUM);
    lat0 = st * (1.0f / SNUM);
  }
  __syncthreads();
  const float km = 111.32f;
  if (tid < SNUM) {
    cx[tid] = (lon[tid] - lon0) * (km * cosf(lat0 * 0.017453292519943295f));
    cy[tid] = (lat[tid] - lat0) * km;
  }
  __syncthreads();
  if (tid == 0) {
    float dsum = 0.f, npos = 0.f;
    for (int i = 0; i < SNUM; ++i)
      for (int j = 0; j < SNUM; ++j) {
        float dx = cx[i] - cx[j], dy = cy[i] - cy[j];
        float d = sqrtf(dx * dx + dy * dy + 1e-12f);
        dsum += d;
        if (d > 1e-6f) npos += 1.f;
      }
    float sc = dsum / fmaxf(npos, 1.f);
    iscale = 1.f / fmaxf(sc, 1e-6f);
  }
  __syncthreads();
  for (int idx = tid; idx < SNUM * SNUM; idx += blockDim.x) {
    int i = idx >> 6, j = idx & 63;
    h_out[2 * idx + 0] = (cx[i] - cx[j]) * iscale;
    h_out[2 * idx + 1] = (cy[i] - cy[j]) * iscale;
  }
}

// -------------------------------------------------------------------------
// 128x128x128 fp32 matmul on the WMMA pipe: D = op(A) * op(B) (+ diag).
// ta: A operand is transposed (read A[k][m]); tb: B operand is transposed
// (read B[n][k]).  8 waves, 64 output tiles of 16x16, K in steps of 4 via
// V_WMMA_F32_16X16X4_F32.  All buffers are LDS, row-major, ld = 128.
// VGPR layouts per CDNA5 ISA 7.12.2:
//   A 16x4 f32 : lane L -> row = L&15, vgpr0/1 hold K = (L>>4)*2 (+1)
//   B 4x16 f32 : lane L -> col = L&15, vgpr0/1 hold K = (L>>4)*2 (+1)
//   C/D 16x16  : lane L -> col = L&15, vgpr r -> row = r + 8*(L>>4)
// -------------------------------------------------------------------------
__device__ __forceinline__ void mm128(const float* __restrict__ A,
                                      const float* __restrict__ B,
                                      float* __restrict__ D,
                                      int tid, int ta, int tb,
                                      float diag_add) {
  const int lane = tid & 31;
  const int wave = tid >> 5;
  const int mr = lane & 15;
  const int kh = (lane >> 4) << 1;  // 0 or 2
  for (int tile = wave; tile < 64; tile += 8) {
    const int ti = tile >> 3, tj = tile & 7;
    v8f acc = {0.f, 0.f, 0.f, 0.f, 0.f, 0.f, 0.f, 0.f};
    for (int k = 0; k < DDIM; k += 4) {
      v2f a, bq;
      if (!ta) {
        const float* ar = A + (ti * 16 + mr) * DDIM + k + kh;
        a.x = ar[0];
        a.y = ar[1];
      } else {
        a.x = A[(k + kh) * DDIM + ti * 16 + mr];
        a.y = A[(k + kh + 1) * DDIM + ti * 16 + mr];
      }
      if (!tb) {
        bq.x = B[(k + kh) * DDIM + tj * 16 + mr];
        bq.y = B[(k + kh + 1) * DDIM + tj * 16 + mr];
      } else {
        const float* br = B + (tj * 16 + mr) * DDIM + k + kh;
        bq.x = br[0];
        bq.y = br[1];
      }
      acc = __builtin_amdgcn_wmma_f32_16x16x4_f32(
          false, a, false, bq, (short)0, acc, false, false);
    }
    const int r0 = ti * 16 + ((lane >> 4) << 3);
    const int c = tj * 16 + mr;
#pragma unroll
    for (int r = 0; r < 8; ++r) {
      const int row = r0 + r;
      float val = acc[r];
      if (row == c) val += diag_add;
      D[row * DDIM + c] = val;
    }
  }
}

// -------------------------------------------------------------------------
// Measurement update: Sm = P + r2*I; Cholesky; X = Sm^{-1} P (-> Ab);
// w = Sm^{-1} v; K = X^T; m += X^T v; P -= (X^T P); returns nll term (tid 0).
// -------------------------------------------------------------------------
__device__ __forceinline__ float kalman_update(
    float* __restrict__ Pb, float* __restrict__ Ab, float* __restrict__ Tb,
    float* __restrict__ Lb, float* __restrict__ mb, float* __restrict__ vb,
    float* __restrict__ wb, float* __restrict__ misc,
    const float* __restrict__ zt, float r2, int tid) {
  if (tid < DDIM) vb[tid] = zt[tid] - mb[tid];
  for (int idx = tid; idx < DDIM * DDIM; idx += NTH) {
    int i = idx >> 7, j = idx & 127;
    Lb[idx] = Pb[idx] + ((i == j) ? r2 : 0.f);
  }
  __syncthreads();
  // Right-looking Cholesky of Sm (lower triangle of Lb).
  for (int k = 0; k < DDIM; ++k) {
    if (tid == 0) {
      float d = sqrtf(fmaxf(Lb[k * DDIM + k], 1e-30f));
      Lb[k * DDIM + k] = d;
      misc[0] = 1.f / d;
    }
    __syncthreads();
    const float inv = misc[0];
    const int i = k + 1 + tid;
    if (i < DDIM) Lb[i * DDIM + k] *= inv;
    __syncthreads();
    if (i < DDIM) {
      const float lik = Lb[i * DDIM + k];
      for (int j = k + 1; j <= i; ++j)
        Lb[i * DDIM + j] -= lik * Lb[j * DDIM + k];
    }
    __syncthreads();
  }
  if (tid == 0) {
    float ld = 0.f;
    for (int i = 0; i < DDIM; ++i) ld += logf(Lb[i * DDIM + i]);
    misc[1] = 2.f * ld;
  }
  // Column-parallel triangular solves: X = Sm^{-1} P (into Ab, A is dead),
  // and w = Sm^{-1} v on thread 128.
  if (tid < DDIM) {
    const int c = tid;
    for (int i = 0; i < DDIM; ++i) {
      float s = Pb[i * DDIM + c];
      for (int j = 0; j < i; ++j) s -= Lb[i * DDIM + j] * Ab[j * DDIM + c];
      Ab[i * DDIM + c] = s / Lb[i * DDIM + i];
    }
    for (int i = DDIM - 1; i >= 0; --i) {
      float s = Ab[i * DDIM + c];
      for (int j = i + 1; j < DDIM; ++j)
        s -= Lb[j * DDIM + i] * Ab[j * DDIM + c];
      Ab[i * DDIM + c] = s / Lb[i * DDIM + i];
    }
  } else if (tid == DDIM) {
    for (int i = 0; i < DDIM; ++i) {
      float s = vb[i];
      for (int j = 0; j < i; ++j) s -= Lb[i * DDIM + j] * wb[j];
      wb[i] = s / Lb[i * DDIM + i];
    }
    for (int i = DDIM - 1; i >= 0; --i) {
      float s = wb[i];
      for (int j = i + 1; j < DDIM; ++j) s -= Lb[j * DDIM + i] * wb[j];
      wb[i] = s / Lb[i * DDIM + i];
    }
  }
  __syncthreads();
  // Tb = K*P = X^T * P  (WMMA, A-operand transposed)
  mm128(Ab, Pb, Tb, tid, 1, 0, 0.f);
  __syncthreads();
  if (tid < DDIM) {
    float s = 0.f;
    for (int k = 0; k < DDIM; ++k) s += Ab[k * DDIM + tid] * vb[k];
    mb[tid] += s;  // m += K v
  }
  for (int idx = tid; idx < DDIM * DDIM; idx += NTH) Pb[idx] -= Tb[idx];
  float nll = 0.f;
  if (tid == 0) {
    float dq = 0.f;
    for (int i = 0; i < DDIM; ++i) dq += vb[i] * wb[i];
    nll = 0.5f * (dq + misc[1] + (float)DDIM * 1.8378770664093453f);
  }
  __syncthreads();
  return nll;
}

// -------------------------------------------------------------------------
// Kernel 1: one workgroup (256 thr = 8 waves) per batch chain. Entire
// working set (P, A/X, temp, L, h) lives in ~292 KB of LDS for all 128 steps.
// -------------------------------------------------------------------------
__global__ void __launch_bounds__(NTH, 1)
ide_kalman_kernel(const float* __restrict__ z, const float* __restrict__ mu,
                  const float* __restrict__ sigma,
                  const float* __restrict__ log_q,
                  const float* __restrict__ log_r,
                  const float* __restrict__ log_p0,
                  const float* __restrict__ log_damp,
                  const float* __restrict__ init_mean,
                  const float* __restrict__ craw,
                  const float* __restrict__ h_glob,
                  float* __restrict__ nll_out) {
  extern __shared__ float smem[];
  float* Pb = smem;                  // 16384  covariance
  float* Ab = Pb + DDIM * DDIM;      // 16384  A, later X = Sm^{-1} P
  float* Tb = Ab + DDIM * DDIM;      // 16384  matmul temp
  float* Lb = Tb + DDIM * DDIM;      // 16384  Sm / Cholesky factor
  float* hb = Lb + DDIM * DDIM;      // 8192   h[i][j][2]
  float* mb = hb + SNUM * SNUM * 2;  // 128    mean
  float* mt = mb + DDIM;             // 128    mean temp
  float* vb = mt + DDIM;             // 128    innovation
  float* wb = vb + DDIM;             // 128    Sm^{-1} v
  float* rs = wb + DDIM;             // 256    kernel row sums
  float* sc = rs + 256;              // 32     per-(t,s) scalars
  float* misc = sc + 32;             // 16

  const int tid = threadIdx.x;
  const int b = blockIdx.x;

  const float r2 = expf(2.f * log_r[0]);
  const float q2 = expf(2.f * log_q[0]);
  const float p0sq = expf(2.f * log_p0[0]);
  const float damping = expf(log_damp[0]);

  // ---- init: P0 = p0^2 I, m0 = init_mean, stage h into LDS ----
  for (int idx = tid; idx < DDIM * DDIM; idx += NTH) {
    int i = idx >> 7, j = idx & 127;
    Pb[idx] = (i == j) ? p0sq : 0.f;
  }
  if (tid < DDIM) mb[tid] = init_mean[tid];
  for (int idx = tid; idx < SNUM * SNUM * 2; idx += NTH) hb[idx] = h_glob[idx];
  __syncthreads();

  float nll_acc =
      kalman_update(Pb, Ab, Tb, Lb, mb, vb, wb, misc,
                    z + (size_t)b * TSTEPS * DDIM, r2, tid);

  for (int t = 1; t < TSTEPS; ++t) {
    const float* zt = z + ((size_t)b * TSTEPS + t) * DDIM;
    if (t + 1 < TSTEPS && tid < 4)
      __builtin_prefetch(z + ((size_t)b * TSTEPS + t + 1) * DDIM + tid * 32, 0, 1);

    // ---- build A from mu_{t-1}, sigma_{t-1} ----
    const float* mu_t = mu + ((size_t)b * (TSTEPS - 1) + (t - 1)) * 4;
    const float* sg_t = sigma + ((size_t)b * (TSTEPS - 1) + (t - 1)) * 16;
    if (tid < 4) {
      const int tt = tid >> 1, ss = tid & 1;
      // SEL[t][s] = al*s0 + be*s1
      const float al = (tt == 0 && ss == 0) ? 1.f : ((tt == 1 && ss == 1) ? 0.f : 0.5f);
      const float be = (tt == 0 && ss == 0) ? 0.f : ((tt == 1 && ss == 1) ? 1.f : 0.5f);
      float S[4][4];
      for (int i = 0; i < 4; ++i)
        for (int j = 0; j < 4; ++j)
          S[i][j] = 0.5f * (sg_t[i * 4 + j] + sg_t[j * 4 + i]);
      const float dm0 = al * mu_t[0] + be * mu_t[2];
      const float dm1 = al * mu_t[1] + be * mu_t[3];
      const float dc00 = al * al * S[0][0] + 2.f * al * be * S[0][2] + be * be * S[2][2];
      const float dc01 = al * al * S[0][1] + al * be * (S[0][3] + S[2][1]) + be * be * S[2][3];
      const float dc11 = al * al * S[1][1] + 2.f * al * be * S[1][3] + be * be * S[3][3];
      const float Dm00 = 1.0001f + 2.f * dc00;
      const float Dm01 = 2.f * dc01;
      const float Dm11 = 1.0001f + 2.f * dc11;
      const float det = Dm00 * Dm11 - Dm01 * Dm01;
      const float inv = 1.f / det;
      sc[tid * 8 + 0] = dm0;
      sc[tid * 8 + 1] = dm1;
      sc[tid * 8 + 2] = Dm11 * inv;   // Dinv00
      sc[tid * 8 + 3] = -Dm01 * inv;  // Dinv01
      sc[tid * 8 + 4] = Dm00 * inv;   // Dinv11
      sc[tid * 8 + 5] = logf(det);
      sc[tid * 8 + 6] =
          ((tt == ss) ? 1.f : 0.f) + 0.25f * tanhf(craw[tt * 2 + ss]);
    }
    __syncthreads();
    // unnormalized kernel -> Ab  (row = tt*64+i, col = ss*64+j)
    for (int idx = tid; idx < 4 * SNUM * SNUM; idx += NTH) {
      const int ts = idx >> 12, rem = idx & 4095;
      const int i = rem >> 6, j = rem & 63;
      const int tt = ts >> 1, ssn = ts & 1;
      const float dx = hb[2 * rem + 0] - sc[ts * 8 + 0];
      const float dy = hb[2 * rem + 1] - sc[ts * 8 + 1];
      const float q = sc[ts * 8 + 2] * dx * dx + 2.f * sc[ts * 8 + 3] * dx * dy +
                      sc[ts * 8 + 4] * dy * dy;
      Ab[(tt * SNUM + i) * DDIM + ssn * SNUM + j] =
          expf(-0.5f * (q + sc[ts * 8 + 5]));
    }
    __syncthreads();
    {  // per-(t,s,i) row sums: exactly 256 rows, one per thread
      const int ts = tid >> 6, i = tid & 63;
      const int tt = ts >> 1, ssn = ts & 1;
      const float* row = Ab + (tt * SNUM + i) * DDIM + ssn * SNUM;
      float s = 0.f;
      for (int j = 0; j < SNUM; ++j) s += row[j];
      rs[tid] = fmaxf(s, 1e-6f);
    }
    __syncthreads();
    // normalize, apply coupling, add (1-damping) on the diagonal
    for (int idx = tid; idx < 4 * SNUM * SNUM; idx += NTH) {
      const int ts = idx >> 12, rem = idx & 4095;
      const int i = rem >> 6, j = rem & 63;
      const int tt = ts >> 1, ssn = ts & 1;
      const int pos = (tt * SNUM + i) * DDIM + ssn * SNUM + j;
      float a = Ab[pos] / rs[ts * 64 + i] * sc[ts * 8 + 6];
      if (tt == ssn && i == j) a += (1.f - damping);
      Ab[pos] = a;
    }
    __syncthreads();

    // ---- predict: m = A m ; P = A P A^T + q2 I ----
    if (tid < DDIM) {
      float s = 0.f;
      for (int k = 0; k < DDIM; ++k) s += Ab[tid * DDIM + k] * mb[k];
      mt[tid] = s;
    }
    mm128(Ab, Pb, Tb, tid, 0, 0, 0.f);  // T = A P
    __syncthreads();
    mm128(Tb, Ab, Pb, tid, 0, 1, q2);   // P = T A^T + q2 I
    if (tid < DDIM) mb[tid] = mt[tid];
    __syncthreads();

    // ---- measurement update ----
    nll_acc += kalman_update(Pb, Ab, Tb, Lb, mb, vb, wb, misc, zt, r2, tid);
  }

  if (tid == 0) nll_out[b] = nll_acc;
}

// -------------------------------------------------------------------------
// Kernel 2: mean over the 16 per-batch NLLs -> scalar output.
// -------------------------------------------------------------------------
__global__ void ide_reduce_kernel(const float* __restrict__ nll,
                                  float* __restrict__ out) {
  if (threadIdx.x == 0) {
    float s = 0.f;
    for (int i = 0; i < BATCH; ++i) s += nll[i];
    out[0] = s * (1.0f / BATCH);
  }
}

extern "C" void kernel_launch(void* const* d_in, const int* in_sizes, int n_in,
                              void* d_out, int out_size, void* d_ws,
                              size_t ws_size, hipStream_t stream) {
  (void)in_sizes; (void)n_in; (void)out_size; (void)ws_size;
  const float* z = (const float*)d_in[0];
  const float* lon = (const float*)d_in[1];
  const float* lat = (const float*)d_in[2];
  const float* mu = (const float*)d_in[3];
  const float* sigma = (const float*)d_in[4];
  const float* lq = (const float*)d_in[5];
  const float* lr = (const float*)d_in[6];
  const float* lp0 = (const float*)d_in[7];
  const float* ldmp = (const float*)d_in[8];
  const float* im = (const float*)d_in[9];
  const float* craw = (const float*)d_in[10];

  float* h_ws = (float*)d_ws;                    // 8192 floats
  float* nll_ws = h_ws + SNUM * SNUM * 2;        // 16 floats

  const size_t smem_bytes =
      (size_t)(4 * DDIM * DDIM + 2 * SNUM * SNUM + 4 * DDIM + 256 + 32 + 16) *
      sizeof(float);  // ~292 KB, fits CDNA5's 320 KB/WGP LDS

  ide_h_kernel<<<1, 256, 0, stream>>>(lon, lat, h_ws);

  (void)hipFuncSetAttribute((const void*)ide_kalman_kernel,
                            hipFuncAttributeMaxDynamicSharedMemorySize,
                            (int)smem_bytes);
  ide_kalman_kernel<<<BATCH, NTH, smem_bytes, stream>>>(
      z, mu, sigma, lq, lr, lp0, ldmp, im, craw, h_ws, nll_ws);

  ide_reduce_kernel<<<1, 32, 0, stream>>>(nll_ws, (float*)d_out);
}